// MultiConvexNet_76209899701135
// MI455X (gfx1250) — compile-verified
//
#include <hip/hip_runtime.h>
#include <math.h>

// ---------------------------------------------------------------------------
// MultiConvexNet forward for MI455X (gfx1250, wave32).
// Transcendental-bound problem: z-GEMM (K=3, zero-padded to K=4) done on the
// matrix pipe with exact-f32 V_WMMA_F32_16X16X4_F32, recomputed per pass so z
// never touches LDS/HBM; pow chains on VALU trans ops (v_log_f32/v_exp_f32).
// A-fragments (centered vertices) are hoisted into 32 VGPRs once per wave so
// the WMMA loops have zero LDS dependence (no s_wait_dscnt stalls).
// LDS arrays are stride-4 padded with a hard 0 in lane 3 so fragment loads
// are unconditional aligned ds_load_b64 (no exec-mask divergence).
// ---------------------------------------------------------------------------

typedef __attribute__((ext_vector_type(2))) float v2f;
typedef __attribute__((ext_vector_type(8))) float v8f;

#define VN 256          // vertices per (b,p)
#define DN 1106         // directions
#define DT 70           // 16-wide direction tiles (padded to 1120)
#define DPAD 1120
#define LBc 1e-20f
#define UBc 1e+20f
#define LB_EXPc -20.0f
#define LOG10_2 0.3010299956639812f   // log10(2)
#define LOG2_10 3.3219280948873623f   // log2(10)

__device__ __forceinline__ float clipLU(float x) {
    return fminf(fmaxf(x, LBc), UBc);
}
__device__ __forceinline__ float flog2(float x) {
    return __builtin_amdgcn_logf(x);       // v_log_f32 (log base 2)
}
__device__ __forceinline__ float fexp2(float x) {
    return __builtin_amdgcn_exp2f(x);      // v_exp_f32 (exp base 2)
}

__global__ void __launch_bounds__(256)
mcnet_kernel(const float* __restrict__ vertices,   // (B,P,V,3)
             const float* __restrict__ smooth,     // (B,P)
             const float* __restrict__ dirs,       // (D,3)
             float* __restrict__ out_points,       // (B,P*D,3)
             float* __restrict__ out_dirh,         // (B,P,D,4)
             float* __restrict__ out_meanv,        // (B,P,3)
             float* __restrict__ out_dirs,         // (D,3)
             float* __restrict__ out_local,        // (B,P,V,3)
             float* __restrict__ out_zeros)        // 2 scalars
{
    __shared__ float lds_local[VN * 4];      // centered vertices, [v*4+c], c=3 -> 0
    __shared__ float lds_dirs[DPAD * 4];     // directions,        [d*4+c], c=3 -> 0
    __shared__ float lds_part[8 * 3];        // per-wave mean partials
    __shared__ float lds_mean[3];

    const int t    = threadIdx.x;            // 0..255
    const int bp   = blockIdx.x;             // 0..255  == b*P + p
    const int lane = t & 31;
    const int wave = t >> 5;
    const float psm = smooth[bp];

    // ---- load this thread's vertex, wave-reduce for the mean -------------
    const float* vbase = vertices + (size_t)bp * VN * 3;
    float vx = vbase[t * 3 + 0];
    float vy = vbase[t * 3 + 1];
    float vz = vbase[t * 3 + 2];

    float sx = vx, sy = vy, sz = vz;
    #pragma unroll
    for (int off = 16; off; off >>= 1) {
        sx += __shfl_xor(sx, off);
        sy += __shfl_xor(sy, off);
        sz += __shfl_xor(sz, off);
    }
    if (lane == 0) {
        lds_part[wave * 3 + 0] = sx;
        lds_part[wave * 3 + 1] = sy;
        lds_part[wave * 3 + 2] = sz;
    }

    // ---- stage directions into LDS, stride 4, zero pad -------------------
    for (int i = t; i < DPAD * 4; i += 256) {
        const int d = i >> 2, c = i & 3;
        lds_dirs[i] = (c < 3 && d < DN) ? dirs[d * 3 + c] : 0.0f;
    }
    __syncthreads();

    if (t < 3) {
        float s = 0.0f;
        #pragma unroll
        for (int w = 0; w < 8; ++w) s += lds_part[w * 3 + t];
        lds_mean[t] = s * (1.0f / VN);
    }
    __syncthreads();

    const float mx = lds_mean[0], my = lds_mean[1], mz = lds_mean[2];
    const float lx = vx - mx, ly = vy - my, lz = vz - mz;
    lds_local[t * 4 + 0] = lx;
    lds_local[t * 4 + 1] = ly;
    lds_local[t * 4 + 2] = lz;
    lds_local[t * 4 + 3] = 0.0f;

    // pass-through outputs
    float* olocal = out_local + (size_t)bp * VN * 3;
    olocal[t * 3 + 0] = lx;
    olocal[t * 3 + 1] = ly;
    olocal[t * 3 + 2] = lz;
    if (t == 0) {
        out_meanv[bp * 3 + 0] = mx;
        out_meanv[bp * 3 + 1] = my;
        out_meanv[bp * 3 + 2] = mz;
    }
    if (bp == 0) {
        for (int i = t; i < DN * 3; i += 256) out_dirs[i] = dirs[i];
        if (t == 0) { out_zeros[0] = 0.0f; out_zeros[1] = 0.0f; }
    }
    __syncthreads();

    // ---- per-wave WMMA fragment indices ----------------------------------
    // A (16x4 f32, M=v, K=c): lanes 0-15 hold K=0,1 ; lanes 16-31 K=2,3
    // B (4x16 f32, K=c, N=d): lane%16 = column d, K = 2*(lane/16)+vgpr
    // C/D (16x16 f32):        lane, vgpr r -> M = r + 8*(lane/16), N = lane%16
    const int half = lane >> 4;
    const int m    = lane & 15;
    const int c0   = 2 * half;

    // Hoist all 16 A-fragments (invariant across tiles & passes): 32 VGPRs.
    v2f afr[16];
    #pragma unroll
    for (int ch = 0; ch < 16; ++ch) {
        const int vrow = ch * 16 + m;
        afr[ch] = *(const v2f*)&lds_local[vrow * 4 + c0];   // aligned b64
    }

    for (int tile = wave; tile < DT; tile += 8) {
        const int d0  = tile * 16;
        const int dme = d0 + m;          // this lane's direction column

        const v2f bfrag = *(const v2f*)&lds_dirs[dme * 4 + c0];  // aligned b64

        // ---------------- pass 1: zmax_d = max_v relu(z) ------------------
        float zmax = 0.0f;
        #pragma unroll
        for (int ch = 0; ch < 16; ++ch) {
            v8f acc = {};
            acc = __builtin_amdgcn_wmma_f32_16x16x4_f32(
                false, afr[ch], false, bfrag, (short)0, acc, false, false);
            #pragma unroll
            for (int r = 0; r < 8; ++r) zmax = fmaxf(zmax, acc[r]);
        }
        zmax = fmaxf(zmax, __shfl_xor(zmax, 16));

        // scale factor k (branch-free; zmax==0 -> k=inf like the reference)
        const float zlog = flog2(zmax) * LOG10_2;            // log10(zmax)
        const float expo = zlog * psm;
        const float lk   = (expo < LB_EXPc) ? (LB_EXPc - expo) / psm : 0.0f;
        const float ck   = fminf(fmaxf(ceilf(lk), 0.0f), UBc);
        const float kk   = fexp2(ck * LOG2_10);              // 10^ck

        // ---------------- pass 2: sum_v clip((zm*k)^p) --------------------
        float ssum = 0.0f;
        for (int ch = 0; ch < 16; ++ch) {
            v8f acc = {};
            acc = __builtin_amdgcn_wmma_f32_16x16x4_f32(
                false, afr[ch], false, bfrag, (short)0, acc, false, false);
            #pragma unroll
            for (int r = 0; r < 8; ++r) {
                const float zms = fmaxf(acc[r], 0.0f) * kk;
                const float zp  = (zms > 0.0f)
                                  ? clipLU(fexp2(psm * flog2(zms)))
                                  : 0.0f;
                ssum += zp;
            }
        }
        ssum += __shfl_xor(ssum, 16);
        const float h    = clipLU(fexp2(flog2(ssum) * (1.0f / psm)));
        const float invh = 1.0f / h;
        const float pm1  = psm - 1.0f;

        // ---------------- pass 3: dhdz + surf accumulation ----------------
        float s0 = 0.0f, s1 = 0.0f, s2 = 0.0f;
        for (int ch = 0; ch < 16; ++ch) {
            v8f acc = {};
            acc = __builtin_amdgcn_wmma_f32_16x16x4_f32(
                false, afr[ch], false, bfrag, (short)0, acc, false, false);
            #pragma unroll
            for (int r = 0; r < 8; ++r) {
                const int v = ch * 16 + r + 8 * half;        // C-layout row
                const float zms   = fmaxf(acc[r], 0.0f) * kk;
                const float ratio = zms * invh;              // NaN if k=inf
                const float dh = (ratio > 0.0f)
                                 ? clipLU(fexp2(pm1 * flog2(ratio)))
                                 : LBc;
                const float4 lv = *(const float4*)&lds_local[v * 4]; // b128
                s0 += dh * lv.x;
                s1 += dh * lv.y;
                s2 += dh * lv.z;
            }
        }
        s0 += __shfl_xor(s0, 16);
        s1 += __shfl_xor(s1, 16);
        s2 += __shfl_xor(s2, 16);

        if (half == 0 && dme < DN) {
            const size_t pbase = ((size_t)bp * DN + dme) * 3;
            out_points[pbase + 0] = s0 + mx;
            out_points[pbase + 1] = s1 + my;
            out_points[pbase + 2] = s2 + mz;
            const size_t dbase = ((size_t)bp * DN + dme) * 4;
            out_dirh[dbase + 0] = lds_dirs[dme * 4 + 0];
            out_dirh[dbase + 1] = lds_dirs[dme * 4 + 1];
            out_dirh[dbase + 2] = lds_dirs[dme * 4 + 2];
            out_dirh[dbase + 3] = h / kk;                    // h/k (0 if k=inf)
        }
    }
}

extern "C" void kernel_launch(void* const* d_in, const int* in_sizes, int n_in,
                              void* d_out, int out_size, void* d_ws, size_t ws_size,
                              hipStream_t stream) {
    const float* vertices = (const float*)d_in[0];   // (8,32,256,3)
    const float* smooth   = (const float*)d_in[1];   // (8,32)
    const float* dirs     = (const float*)d_in[2];   // (1106,3)

    float* out        = (float*)d_out;
    float* out_points = out;                                  // 849408
    float* out_dirh   = out_points + 849408;                  // 1132544
    float* out_meanv  = out_dirh + 1132544;                   // 768
    float* out_dirs   = out_meanv + 768;                      // 3318
    float* out_local  = out_dirs + 3318;                      // 196608
    float* out_zeros  = out_local + 196608;                   // 2

    mcnet_kernel<<<256, 256, 0, stream>>>(vertices, smooth, dirs,
                                          out_points, out_dirh, out_meanv,
                                          out_dirs, out_local, out_zeros);
}